// AttentionConv1d_54322746360501
// MI455X (gfx1250) — compile-verified
//
#include <hip/hip_runtime.h>
#include <math.h>

typedef __attribute__((ext_vector_type(2))) float v2f;
typedef __attribute__((ext_vector_type(8))) float v8f;

#define B_    32
#define CIN   128
#define NLEN  4096
#define SBINS 2049
#define SP    2064              // padded spectrum stride (64B-aligned rows)
#define NTOK  (B_ * SBINS)      // 65568 = 16 * 4098
#define NTILE (NTOK / 16)       // 4098

// ---------------------------------------------------------------------------
// Forward FFT: one 4096-pt complex FFT (real input) per (b,c) row.
// ---------------------------------------------------------------------------
__global__ __launch_bounds__(256) void kfft_fwd(const float* __restrict__ x,
                                                float* __restrict__ xr,
                                                float* __restrict__ xi) {
  __shared__ float sr[NLEN];
  __shared__ float si[NLEN];
  const int row = blockIdx.x;            // b*128 + c
  const int tid = threadIdx.x;
  const float* xrow = x + (size_t)row * NLEN;
  for (int k = tid; k < NLEN; k += 256) {
    int rv = (int)(__brev((unsigned)k) >> 20);   // 12-bit bit reverse
    sr[rv] = xrow[k];
    si[rv] = 0.0f;
  }
  __syncthreads();
  for (int s = 1; s <= 12; ++s) {
    const int m = 1 << s;
    const int half = m >> 1;
    const float ang0 = -6.283185307179586f / (float)m;
    for (int idx = tid; idx < (NLEN / 2); idx += 256) {
      const int grp = idx >> (s - 1);
      const int pos = idx & (half - 1);
      const int base = (grp << s) + pos;
      float wi_s, wr_c;
      __sincosf(ang0 * (float)pos, &wi_s, &wr_c);
      const float ar = sr[base], ai = si[base];
      const float br = sr[base + half], bi = si[base + half];
      const float tr = wr_c * br - wi_s * bi;
      const float ti = wr_c * bi + wi_s * br;
      sr[base] = ar + tr;        si[base] = ai + ti;
      sr[base + half] = ar - tr; si[base + half] = ai - ti;
    }
    __syncthreads();
  }
  float* orr = xr + (size_t)row * SP;
  float* ori = xi + (size_t)row * SP;
  for (int s = tid; s < SBINS; s += 256) { orr[s] = sr[s]; ori[s] = si[s]; }
}

// ---------------------------------------------------------------------------
// Inverse FFT: Hermitian-extend spectrum, 4096-pt IFFT, real part / N.
// ---------------------------------------------------------------------------
__global__ __launch_bounds__(256) void kfft_inv(const float* __restrict__ yr,
                                                const float* __restrict__ yi,
                                                float* __restrict__ out) {
  __shared__ float sr[NLEN];
  __shared__ float si[NLEN];
  const int row = blockIdx.x;
  const int tid = threadIdx.x;
  const float* rrow = yr + (size_t)row * SP;
  const float* irow = yi + (size_t)row * SP;
  for (int k = tid; k < NLEN; k += 256) {
    float vr, vi;
    if (k <= 2048) { vr = rrow[k];        vi = irow[k]; }
    else           { vr = rrow[NLEN - k]; vi = -irow[NLEN - k]; }
    int rv = (int)(__brev((unsigned)k) >> 20);
    sr[rv] = vr; si[rv] = vi;
  }
  __syncthreads();
  for (int s = 1; s <= 12; ++s) {
    const int m = 1 << s;
    const int half = m >> 1;
    const float ang0 = 6.283185307179586f / (float)m;   // +i for inverse
    for (int idx = tid; idx < (NLEN / 2); idx += 256) {
      const int grp = idx >> (s - 1);
      const int pos = idx & (half - 1);
      const int base = (grp << s) + pos;
      float wi_s, wr_c;
      __sincosf(ang0 * (float)pos, &wi_s, &wr_c);
      const float ar = sr[base], ai = si[base];
      const float br = sr[base + half], bi = si[base + half];
      const float tr = wr_c * br - wi_s * bi;
      const float ti = wr_c * bi + wi_s * br;
      sr[base] = ar + tr;        si[base] = ai + ti;
      sr[base + half] = ar - tr; si[base + half] = ai - ti;
    }
    __syncthreads();
  }
  float* orow = out + (size_t)row * NLEN;
  const float inv = 1.0f / (float)NLEN;
  for (int k = tid; k < NLEN; k += 256) orow[k] = sr[k] * inv;
}

// ---------------------------------------------------------------------------
// Weight fusion kernels (complex inputs are interleaved re/im float pairs)
// ---------------------------------------------------------------------------
__global__ __launch_bounds__(256) void kprep_wcomb(const float* __restrict__ pw,
                                                   const float* __restrict__ ow,
                                                   float* __restrict__ wc) {
  int idx = blockIdx.x * 256 + threadIdx.x;      // i*256 + o
  if (idx >= 128 * 256) return;
  int i = idx >> 8, o = idx & 255;
  float ar = 0.f, ai = 0.f;
  for (int m = 0; m < 128; ++m) {
    float pr = pw[2 * (i * 128 + m)], pi = pw[2 * (i * 128 + m) + 1];
    float qr = ow[2 * (m * 256 + o)], qi = ow[2 * (m * 256 + o) + 1];
    ar += pr * qr - pi * qi;
    ai += pr * qi + pi * qr;
  }
  wc[2 * idx] = ar; wc[2 * idx + 1] = ai;
}

__global__ __launch_bounds__(256) void kprep_m(const float* __restrict__ qw,
                                               const float* __restrict__ kw,
                                               float* __restrict__ btmr,
                                               float* __restrict__ btmi) {
  int idx = blockIdx.x * 256 + threadIdx.x;      // i*128 + j
  if (idx >= 16384) return;
  int i = idx >> 7, j = idx & 127;
  float ar = 0.f, ai = 0.f;
  for (int o = 0; o < 256; ++o) {
    float qr = qw[2 * (o * 128 + i)], qi = qw[2 * (o * 128 + i) + 1];
    float kr = kw[2 * (o * 128 + j)], ki = kw[2 * (o * 128 + j) + 1];
    ar += qr * kr - qi * ki;
    ai += qr * ki + qi * kr;
  }
  btmr[j * 128 + i] = ar;   // stored transposed: [K=j][N=i]
  btmi[j * 128 + i] = ai;
}

__global__ __launch_bounds__(256) void kprep_w2(const float* __restrict__ wc,
                                                const float* __restrict__ vw,
                                                float* __restrict__ btw2r,
                                                float* __restrict__ btw2i) {
  int idx = blockIdx.x * 256 + threadIdx.x;      // i*128 + j
  if (idx >= 16384) return;
  int i = idx >> 7, j = idx & 127;
  float ar = 0.f, ai = 0.f;
  for (int o = 0; o < 256; ++o) {
    float wr = wc[2 * (i * 256 + o)], wi = wc[2 * (i * 256 + o) + 1];
    float vr = vw[2 * (o * 128 + j)], vi = vw[2 * (o * 128 + j) + 1];
    ar += wr * vr - wi * vi;
    ai += wr * vi + wi * vr;
  }
  btw2r[j * 128 + i] = ar;
  btw2i[j * 128 + i] = ai;
}

__global__ __launch_bounds__(128) void kprep_vec(
    const float* __restrict__ qw, const float* __restrict__ qb,
    const float* __restrict__ kw, const float* __restrict__ kb,
    const float* __restrict__ vb, const float* __restrict__ ob,
    const float* __restrict__ pw, const float* __restrict__ pb,
    const float* __restrict__ wc,
    float* __restrict__ rr, float* __restrict__ ri,
    float* __restrict__ c2r, float* __restrict__ c2i,
    float* __restrict__ bcr, float* __restrict__ bci,
    float* __restrict__ c0) {
  int j = threadIdx.x;
  if (j >= 128) return;
  // r = Wk^T bq + Wq^T bk
  float ar = 0.f, ai = 0.f;
  for (int o = 0; o < 256; ++o) {
    float kwr = kw[2 * (o * 128 + j)], kwi = kw[2 * (o * 128 + j) + 1];
    float qbr = qb[2 * o], qbi = qb[2 * o + 1];
    float qwr = qw[2 * (o * 128 + j)], qwi = qw[2 * (o * 128 + j) + 1];
    float kbr = kb[2 * o], kbi = kb[2 * o + 1];
    ar += kwr * qbr - kwi * qbi + qwr * kbr - qwi * kbi;
    ai += kwr * qbi + kwi * qbr + qwr * kbi + qwi * kbr;
  }
  rr[j] = ar; ri[j] = ai;
  // c2 = Wcomb bv
  float cr = 0.f, ci = 0.f;
  for (int o = 0; o < 256; ++o) {
    float wr = wc[2 * (j * 256 + o)], wi = wc[2 * (j * 256 + o) + 1];
    float vr = vb[2 * o], vi = vb[2 * o + 1];
    cr += wr * vr - wi * vi;
    ci += wr * vi + wi * vr;
  }
  c2r[j] = cr; c2i[j] = ci;
  // bcomb = Wproj bout + bproj
  float br = pb[2 * j], bi = pb[2 * j + 1];
  for (int m = 0; m < 128; ++m) {
    float pr = pw[2 * (j * 128 + m)], pim = pw[2 * (j * 128 + m) + 1];
    float obr = ob[2 * m], obi = ob[2 * m + 1];
    br += pr * obr - pim * obi;
    bi += pr * obi + pim * obr;
  }
  bcr[j] = br; bci[j] = bi;
  if (j == 0) {
    float r0 = 0.f, i0 = 0.f;
    for (int o = 0; o < 256; ++o) {
      float a = qb[2 * o], b = qb[2 * o + 1];
      float c = kb[2 * o], d = kb[2 * o + 1];
      r0 += a * c - b * d;
      i0 += a * d + b * c;
    }
    c0[0] = r0; c0[1] = i0;
  }
}

// ---------------------------------------------------------------------------
// Fused attention kernel: both complex GEMMs (XM = X M^T and Z = X W2^T)
// share A fragments; 8 f32 WMMA chains per K-step.  Then per-token phase
// p = sum_i x_i (XM_i + r_i) + c0, phase = p/|p| (LDS), and the output
// spectrum y = phase*(Z + c2) + bc is written in place over X.
// Tile: 16 tokens, 8 waves x 16 outputs, K = 128.
// ---------------------------------------------------------------------------
__global__ __launch_bounds__(256) void kattn(
    float* __restrict__ xr, float* __restrict__ xi,
    const float* __restrict__ btmr, const float* __restrict__ btmi,
    const float* __restrict__ btw2r, const float* __restrict__ btw2i,
    const float* __restrict__ rr, const float* __restrict__ ri,
    const float* __restrict__ c2r, const float* __restrict__ c2i,
    const float* __restrict__ bcr, const float* __restrict__ bci,
    const float* __restrict__ c0) {
  __shared__ float part[8][16][2];
  __shared__ float ph[16][2];
  __shared__ float ytr[128 * 17];
  __shared__ float yti[128 * 17];
  const int tile = blockIdx.x;
  const int tid = threadIdx.x;
  const int w = tid >> 5;         // wave id -> N tile
  const int lane = tid & 31;
  const int hi = lane >> 4;
  const int lo = lane & 15;
  const int t0 = tile * 16;
  // A-fragment role: this lane feeds token t0+lo
  const int tA = t0 + lo;
  const int bA = tA / SBINS;
  const int sA = tA - bA * SBINS;
  const int iN = w * 16 + lo;     // B-fragment / D column index
  v8f aP = {0.f,0.f,0.f,0.f,0.f,0.f,0.f,0.f};
  v8f aQ = aP, aR = aP, aS = aP;              // M-GEMM accumulators
  v8f bP = aP, bQ = aP, bR = aP, bS = aP;     // W2-GEMM accumulators
  for (int j0 = 0; j0 < 128; j0 += 4) {
    const int jA = j0 + 2 * hi;
    const size_t a0 = ((size_t)(bA * CIN + jA)) * SP + sA;
    v2f axr, axi, m_r, m_i, w_r, w_i;
    axr.x = xr[a0];      axr.y = xr[a0 + SP];
    axi.x = xi[a0];      axi.y = xi[a0 + SP];
    m_r.x = btmr[jA * 128 + iN];  m_r.y = btmr[(jA + 1) * 128 + iN];
    m_i.x = btmi[jA * 128 + iN];  m_i.y = btmi[(jA + 1) * 128 + iN];
    w_r.x = btw2r[jA * 128 + iN]; w_r.y = btw2r[(jA + 1) * 128 + iN];
    w_i.x = btw2i[jA * 128 + iN]; w_i.y = btw2i[(jA + 1) * 128 + iN];
    if (j0 + 4 < 128) {                 // prefetch next K-block of X rows
      __builtin_prefetch(&xr[a0 + 4 * SP], 0, 0);
      __builtin_prefetch(&xi[a0 + 4 * SP], 0, 0);
    }
    aP = __builtin_amdgcn_wmma_f32_16x16x4_f32(false, axr, false, m_r, (short)0, aP, false, false);
    aQ = __builtin_amdgcn_wmma_f32_16x16x4_f32(false, axi, false, m_i, (short)0, aQ, false, false);
    aR = __builtin_amdgcn_wmma_f32_16x16x4_f32(false, axr, false, m_i, (short)0, aR, false, false);
    aS = __builtin_amdgcn_wmma_f32_16x16x4_f32(false, axi, false, m_r, (short)0, aS, false, false);
    bP = __builtin_amdgcn_wmma_f32_16x16x4_f32(false, axr, false, w_r, (short)0, bP, false, false);
    bQ = __builtin_amdgcn_wmma_f32_16x16x4_f32(false, axi, false, w_i, (short)0, bQ, false, false);
    bR = __builtin_amdgcn_wmma_f32_16x16x4_f32(false, axr, false, w_i, (short)0, bR, false, false);
    bS = __builtin_amdgcn_wmma_f32_16x16x4_f32(false, axi, false, w_r, (short)0, bS, false, false);
  }
  // ---- stage 1: per-token phase ----
  const float rvr = rr[iN], rvi = ri[iN];
#pragma unroll
  for (int rg = 0; rg < 8; ++rg) {
    const int tl = rg + 8 * hi;          // token of D row rg for this lane
    const int t = t0 + tl;
    const int b = t / SBINS;
    const int s = t - b * SBINS;
    const size_t xa = ((size_t)(b * CIN + iN)) * SP + s;
    const float xvr = xr[xa], xvi = xi[xa];
    const float XMr = aP[rg] - aQ[rg] + rvr;
    const float XMi = aR[rg] + aS[rg] + rvi;
    float vr = xvr * XMr - xvi * XMi;
    float vi = xvr * XMi + xvi * XMr;
    for (int m = 1; m < 16; m <<= 1) {
      vr += __shfl_xor(vr, m, 32);
      vi += __shfl_xor(vi, m, 32);
    }
    if (lo == 0) { part[w][tl][0] = vr; part[w][tl][1] = vi; }
  }
  __syncthreads();
  if (tid < 16) {
    float pr = c0[0], pi = c0[1];
    for (int ww = 0; ww < 8; ++ww) { pr += part[ww][tid][0]; pi += part[ww][tid][1]; }
    const float inv = rsqrtf(pr * pr + pi * pi + 1e-38f);
    ph[tid][0] = pr * inv;
    ph[tid][1] = pi * inv;
  }
  __syncthreads();
  // ---- stage 2: apply phase, write output spectrum (in place) ----
  const float cvr = c2r[iN], cvi = c2i[iN];
  const float bvr = bcr[iN], bvi = bci[iN];
#pragma unroll
  for (int rg = 0; rg < 8; ++rg) {
    const int tl = rg + 8 * hi;
    const float zr = bP[rg] - bQ[rg] + cvr;
    const float zi = bR[rg] + bS[rg] + cvi;
    const float phr = ph[tl][0];
    const float phi = ph[tl][1];
    ytr[iN * 17 + tl] = phr * zr - phi * zi + bvr;
    yti[iN * 17 + tl] = phr * zi + phi * zr + bvi;
  }
  __syncthreads();
  for (int idx = tid; idx < 2048; idx += 256) {
    const int i = idx >> 4;
    const int tl = idx & 15;
    const int t = t0 + tl;
    const int b = t / SBINS;
    const int s = t - b * SBINS;
    const size_t oa = ((size_t)(b * CIN + i)) * SP + s;
    xr[oa] = ytr[i * 17 + tl];
    xi[oa] = yti[i * 17 + tl];
  }
}

// ---------------------------------------------------------------------------
extern "C" void kernel_launch(void* const* d_in, const int* in_sizes, int n_in,
                              void* d_out, int out_size, void* d_ws, size_t ws_size,
                              hipStream_t stream) {
  const float* x  = (const float*)d_in[0];
  const float* qw = (const float*)d_in[1];
  const float* qb = (const float*)d_in[2];
  const float* kw = (const float*)d_in[3];
  const float* kb = (const float*)d_in[4];
  const float* vw = (const float*)d_in[5];
  const float* vb = (const float*)d_in[6];
  const float* ow = (const float*)d_in[7];
  const float* ob = (const float*)d_in[8];
  const float* pw = (const float*)d_in[9];
  const float* pb = (const float*)d_in[10];

  float* ws = (float*)d_ws;
  size_t o = 0;
  float* XR    = ws + o; o += (size_t)B_ * CIN * SP;   // 8,454,144 floats
  float* XI    = ws + o; o += (size_t)B_ * CIN * SP;
  float* BTMR  = ws + o; o += 16384;
  float* BTMI  = ws + o; o += 16384;
  float* BTW2R = ws + o; o += 16384;
  float* BTW2I = ws + o; o += 16384;
  float* WC    = ws + o; o += 65536;                    // Wcomb interleaved
  float* RR    = ws + o; o += 128;
  float* RI    = ws + o; o += 128;
  float* C2R   = ws + o; o += 128;
  float* C2I   = ws + o; o += 128;
  float* BCR   = ws + o; o += 128;
  float* BCI   = ws + o; o += 128;
  float* C0    = ws + o; o += 128;

  kprep_wcomb<<<128, 256, 0, stream>>>(pw, ow, WC);
  kprep_m<<<64, 256, 0, stream>>>(qw, kw, BTMR, BTMI);
  kprep_w2<<<64, 256, 0, stream>>>(WC, vw, BTW2R, BTW2I);
  kprep_vec<<<1, 128, 0, stream>>>(qw, qb, kw, kb, vb, ob, pw, pb, WC,
                                   RR, RI, C2R, C2I, BCR, BCI, C0);
  kfft_fwd<<<B_ * CIN, 256, 0, stream>>>(x, XR, XI);
  kattn<<<NTILE, 256, 0, stream>>>(XR, XI, BTMR, BTMI, BTW2R, BTW2I,
                                   RR, RI, C2R, C2I, BCR, BCI, C0);
  kfft_inv<<<B_ * CIN, 256, 0, stream>>>(XR, XI, (float*)d_out);
}